// TALE_29317446763356
// MI455X (gfx1250) — compile-verified
//
#include <hip/hip_runtime.h>
#include <math.h>

// Problem constants (match reference)
#define BB 8192
#define CC 32
#define RR 24
#define EE 128
#define NSLICE 24

// LDS row stride for the staged B tile: 130 floats.
// 130 % 64 == 2  -> the 16 rows of a B fragment land on distinct bank pairs
// (stride 128 would put every row on the same bank). 130*4 bytes is 8B-aligned,
// so ds_load_b64 of {kbase, kbase+1} (kbase even) stays aligned.
#define LDSS 130

typedef __attribute__((ext_vector_type(2))) float v2f;
typedef __attribute__((ext_vector_type(8))) float v8f;

// ---------------------------------------------------------------------------
// Wave32 butterfly reductions (warpSize == 32 on gfx1250)
// ---------------------------------------------------------------------------
__device__ __forceinline__ float wave_sum(float v) {
    #pragma unroll
    for (int off = 16; off > 0; off >>= 1) v += __shfl_xor(v, off, 32);
    return v;
}
__device__ __forceinline__ float wave_max(float v) {
    #pragma unroll
    for (int off = 16; off > 0; off >>= 1) v = fmaxf(v, __shfl_xor(v, off, 32));
    return v;
}

// ---------------------------------------------------------------------------
// Kernel 1: ctx[b,:] = sum_c input_embed[context[b,c], :]
// grid = B blocks, block = 128 threads (one thread per embedding element).
// Each gathered row is a 512B fully-coalesced load across the block.
// ---------------------------------------------------------------------------
__global__ void ctx_kernel(const int* __restrict__ context,
                           const float* __restrict__ input_embed,
                           float* __restrict__ ctx) {
    const int b = blockIdx.x;
    const int e = threadIdx.x;
    float acc = 0.0f;
    #pragma unroll 4
    for (int c = 0; c < CC; ++c) {
        const int loc = context[b * CC + c];          // uniform per block -> scalar load
        acc += input_embed[(long)loc * EE + e];       // 512B coalesced row gather
    }
    ctx[(long)b * EE + e] = acc;
}

// ---------------------------------------------------------------------------
// Kernel 2: slice logits via V_WMMA_F32_16X16X4_F32.
// D(16x16) = A(16x4 ctx chunk) * B(4x16 slice-embed chunk), K-loop over E=128.
// Each wave owns a 16(M batch rows) x 32(N slices, padded from 24) tile.
// A layout (ISA 7.12.2): lanes 0-15 hold K0/K1 in v0/v1, lanes 16-31 hold K2/K3.
// B layout mirrors C/D striping: v0 = rows K0|K2, v1 = rows K1|K3 per lane half.
// The B operand (24x128 slice embeds + 8 zero rows) is staged once per block
// into LDS, pre-padded -> the inner loop is branch-free (no EXEC juggling
// around the WMMAs) and all 8 waves share one copy.
// grid = B/(16*8) = 64 blocks, block = 256 threads (8 waves).
// ---------------------------------------------------------------------------
__global__ void slice_wmma_kernel(const float* __restrict__ ctx,
                                  const float* __restrict__ slice_embed,
                                  float* __restrict__ logits) {
    __shared__ float sB[32 * LDSS];

    // Cooperative staged load of padded B: rows 0..23 = slice_embed, 24..31 = 0.
    for (int i = threadIdx.x; i < 32 * EE; i += 256) {
        const int r = i >> 7;          // row   (i / 128)
        const int c = i & (EE - 1);    // col   (i % 128)
        sB[r * LDSS + c] = (r < NSLICE) ? slice_embed[i] : 0.0f;
    }
    __syncthreads();

    const int lane = threadIdx.x & 31;
    const int wave = threadIdx.x >> 5;
    const int rowbase = (blockIdx.x * 8 + wave) * 16;   // 16 batch rows per wave

    const int half = lane >> 4;       // 0: lanes 0-15, 1: lanes 16-31
    const int l    = lane & 15;

    const float* arowp = ctx + (long)(rowbase + l) * EE + half * 2;
    const float* bp0   = sB + l * LDSS + half * 2;          // N-tile 0: slices 0..15
    const float* bp1   = sB + (16 + l) * LDSS + half * 2;   // N-tile 1: slices 16..31

    v8f acc0 = {};
    v8f acc1 = {};

    #pragma unroll 4
    for (int kk = 0; kk < EE / 4; ++kk) {
        const int k4 = kk * 4;

        v2f a;                        // A fragment: ctx rows (global, coalesced b64)
        a.x = arowp[k4 + 0];
        a.y = arowp[k4 + 1];

        v2f b0;                       // B fragments from LDS (ds_load_b64, no conflicts)
        b0.x = bp0[k4 + 0];
        b0.y = bp0[k4 + 1];

        v2f b1;
        b1.x = bp1[k4 + 0];
        b1.y = bp1[k4 + 1];

        acc0 = __builtin_amdgcn_wmma_f32_16x16x4_f32(
            false, a, false, b0, (short)0, acc0, false, false);
        acc1 = __builtin_amdgcn_wmma_f32_16x16x4_f32(
            false, a, false, b1, (short)0, acc1, false, false);
    }

    // D layout: VGPR v, lanes 0-15 -> M=v, lanes 16-31 -> M=v+8, N=lane&15
    #pragma unroll
    for (int v = 0; v < 8; ++v) {
        const int row = rowbase + v + half * 8;
        logits[(long)row * 32 + l]      = acc0[v];
        logits[(long)row * 32 + 16 + l] = acc1[v];
    }
}

// ---------------------------------------------------------------------------
// Kernel 3: per-row softmax pick + 24 gathered route dots -> pr[b].
// One wave per batch row; 8 rows per 256-thread block. grid = B/8 blocks.
// ---------------------------------------------------------------------------
__global__ void route_kernel(const int* __restrict__ route,
                             const int* __restrict__ lr,
                             const int* __restrict__ slice_idx,
                             const float* __restrict__ inner_embed,
                             const float* __restrict__ ctx,
                             const float* __restrict__ logits,
                             float* __restrict__ pr) {
    const int lane = threadIdx.x & 31;
    const int wave = threadIdx.x >> 5;
    const int b = blockIdx.x * 8 + wave;

    // ---- slice probability via wave-resident softmax over 24 logits ----
    const float lg = (lane < NSLICE) ? logits[(long)b * 32 + lane] : -INFINITY;
    const float mx = wave_max(lg);
    const float ex = (lane < NSLICE) ? __expf(lg - mx) : 0.0f;
    const float denom = wave_sum(ex);
    const int   sidx = slice_idx[b];                     // uniform per wave
    const float lg_t = __shfl(lg, sidx, 32);
    const float slice_pre = __expf(lg_t - mx) / denom;

    // ---- cache this row's ctx: 4 elements per lane ----
    const float* crow = ctx + (long)b * EE;
    const float c0 = crow[lane];
    const float c1 = crow[lane + 32];
    const float c2 = crow[lane + 64];
    const float c3 = crow[lane + 96];

    // ---- 24 gathered dot products + sigmoid product ----
    float prod = 1.0f;
    #pragma unroll 4
    for (int r = 0; r < RR; ++r) {
        const int ridx = route[b * RR + r];              // uniform per wave
        const int lrv  = lr[b * RR + r];
        const float* irow = inner_embed + (long)ridx * EE;
        float d = irow[lane] * c0
                + irow[lane + 32] * c1
                + irow[lane + 64] * c2
                + irow[lane + 96] * c3;
        d = wave_sum(d);                                 // all lanes get the dot
        const float p = 1.0f / (1.0f + __expf(-d));
        float h = lrv ? p : (1.0f - p);                  // (2p)^lr - p
        if (ridx == 0) h = 1.0f;                         // padded route position
        prod *= h;
    }

    if (lane == 0) pr[b] = slice_pre * prod;
}

// ---------------------------------------------------------------------------
// Kernel 4: deterministic single-block reduction -> 1 - mean(pr)
// ---------------------------------------------------------------------------
__global__ void finalize_kernel(const float* __restrict__ pr, float* __restrict__ out) {
    __shared__ float smem[256];
    const int tid = threadIdx.x;
    float s = 0.0f;
    for (int i = tid; i < BB; i += 256) s += pr[i];
    smem[tid] = s;
    __syncthreads();
    for (int stride = 128; stride > 0; stride >>= 1) {
        if (tid < stride) smem[tid] += smem[tid + stride];
        __syncthreads();
    }
    if (tid == 0) out[0] = 1.0f - smem[0] / (float)BB;
}

// ---------------------------------------------------------------------------
extern "C" void kernel_launch(void* const* d_in, const int* in_sizes, int n_in,
                              void* d_out, int out_size, void* d_ws, size_t ws_size,
                              hipStream_t stream) {
    const int*   context     = (const int*)d_in[0];   // (B, C)
    const int*   route       = (const int*)d_in[1];   // (B, R)
    const int*   lr          = (const int*)d_in[2];   // (B, R)
    const int*   slice_idx   = (const int*)d_in[3];   // (B,)
    const float* input_embed = (const float*)d_in[4]; // (NUM_LOC, E)
    const float* inner_embed = (const float*)d_in[5]; // (NUM_INNER+1, E)
    const float* slice_embed = (const float*)d_in[6]; // (NUM_SLICES, E)
    float* out = (float*)d_out;

    // Workspace layout (floats): ctx[B*E] | logits[B*32] | pr[B]
    float* ctx    = (float*)d_ws;
    float* logits = ctx + (long)BB * EE;
    float* pr     = logits + (long)BB * 32;

    ctx_kernel<<<BB, EE, 0, stream>>>(context, input_embed, ctx);
    slice_wmma_kernel<<<BB / (16 * 8), 256, 0, stream>>>(ctx, slice_embed, logits);
    route_kernel<<<BB / 8, 256, 0, stream>>>(route, lr, slice_idx, inner_embed,
                                             ctx, logits, pr);
    finalize_kernel<<<1, 256, 0, stream>>>(pr, out);
}